// LSTM_61220463837402
// MI455X (gfx1250) — compile-verified
//
#include <hip/hip_runtime.h>
#include <hip/hip_bf16.h>

// Problem constants (from reference)
#define Bb   16384
#define Tt   90
#define Hh   8
#define OUTL 30

typedef float v2f __attribute__((ext_vector_type(2)));
typedef float v8f __attribute__((ext_vector_type(8)));

// D = A(16x4 f32) * B(4x16 f32) + C, full fp32 WMMA
__device__ __forceinline__ v8f wmma4(float a0, float a1, float b0, float b1, v8f c) {
  v2f a = {a0, a1};
  v2f b = {b0, b1};
  return __builtin_amdgcn_wmma_f32_16x16x4_f32(false, a, false, b, (short)0, c, false, false);
}

// K=8 matmul tile: two chained K=4 WMMAs. a[4] = {c0v0,c0v1,c1v0,c1v1}, b[4] same.
__device__ __forceinline__ v8f mm8(const float* a, const float* b, v8f c) {
  c = wmma4(a[2], a[3], b[2], b[3], c);  // K = 4..7
  c = wmma4(a[0], a[1], b[0], b[1], c);  // K = 0..3
  return c;
}

__device__ __forceinline__ float sigm(float x) {
  return __builtin_amdgcn_rcpf(1.0f + __expf(-x));
}
__device__ __forceinline__ float tanh_(float x) {
  return 2.0f * __builtin_amdgcn_rcpf(1.0f + __expf(-2.0f * x)) - 1.0f;
}

// LSTM cell update from gate tiles.
// d0: VGPR r, lanes<16 -> i_r[batch], lanes>=16 -> f_r[batch]
// d1: VGPR r, lanes<16 -> g_r[batch], lanes>=16 -> o_r[batch]
// C (c-state) canonical in lanes<16. Bop rebuilt as B-operand of new h.
// tanh(x) = 2*sigmoid(2x)-1: one sigmoid serves both lane halves of d1.
__device__ __forceinline__ void cell(v8f d0, v8f d1, bool lo,
                                     float* C, float* Bop, float* Hout) {
  float Hl[8];
  const float amul = lo ? 2.0f : 1.0f;   // pre-scale for tanh-via-sigmoid (lo half)
  const float gadd = lo ? -1.0f : 0.0f;  // post-fma: lo -> 2s-1, hi -> s
#pragma unroll
  for (int r = 0; r < 8; r++) {
    float g0 = sigm(d0[r]);                       // I (lo) / F (hi)
    float s  = sigm(amul * d1[r]);
    float g1 = __builtin_fmaf(amul, s, gadd);     // G=tanh (lo) / O=sigm (hi)
    float F = __shfl_xor(g0, 16, 32);             // lanes<16 receive F
    float O = __shfl_xor(g1, 16, 32);             // lanes<16 receive O
    float cn = F * C[r] + g0 * g1;                // valid in lanes<16: f*c + i*g
    C[r] = cn;
    Hl[r] = O * tanh_(cn);                        // valid in lanes<16
  }
  // Build B-operand (h^T, K=8, N=16): VGPR {c,v}: lanes<16 hold k=4c+v, lanes>=16 k=4c+v+2
  float s2 = __shfl_xor(Hl[2], 16, 32);
  float s3 = __shfl_xor(Hl[3], 16, 32);
  float s6 = __shfl_xor(Hl[6], 16, 32);
  float s7 = __shfl_xor(Hl[7], 16, 32);
  Bop[0] = lo ? Hl[0] : s2;
  Bop[1] = lo ? Hl[1] : s3;
  Bop[2] = lo ? Hl[4] : s6;
  Bop[3] = lo ? Hl[5] : s7;
#pragma unroll
  for (int r = 0; r < 8; r++) Hout[r] = Hl[r];
}

__global__ __launch_bounds__(256) void lstm3_kernel(
    const float* __restrict__ x,
    const float* __restrict__ Wih0, const float* __restrict__ Whh0,
    const float* __restrict__ bih0, const float* __restrict__ bhh0,
    const float* __restrict__ Wih1, const float* __restrict__ Whh1,
    const float* __restrict__ bih1, const float* __restrict__ bhh1,
    const float* __restrict__ Wih2, const float* __restrict__ Whh2,
    const float* __restrict__ bih2, const float* __restrict__ bhh2,
    float* __restrict__ out) {
  const int lane = threadIdx.x & 31;
  const int wave = threadIdx.x >> 5;
  const int tile = blockIdx.x * 8 + wave;   // 0..1023
  const int b0 = tile * 16;
  const bool lo = lane < 16;
  const int bl = lane & 15;
  const int kofs = lo ? 0 : 2;
  const int go = lo ? 0 : 8;

  // --- A-operands: W[32x8] -> 2 M-tiles x 2 K-chunks x 2 VGPRs, loaded once ---
  float aH0[8], aH1[8], aH2[8], aI1[8], aI2[8];
  const float* Wm[5] = {Whh0, Whh1, Whh2, Wih1, Wih2};
  float* Am[5] = {aH0, aH1, aH2, aI1, aI2};
#pragma unroll
  for (int w = 0; w < 5; w++)
#pragma unroll
    for (int t = 0; t < 2; t++)
#pragma unroll
      for (int c = 0; c < 2; c++)
#pragma unroll
        for (int v = 0; v < 2; v++)
          Am[w][t * 4 + c * 2 + v] = Wm[w][(t * 16 + bl) * 8 + (c * 4 + v + kofs)];

  // --- biases (b_ih + b_hh), in D layout: VGPR r -> gate (tile*16 + r + go) ---
  float bs0t0[8], bs0t1[8], bs1t0[8], bs1t1[8], bs2t0[8], bs2t1[8];
#pragma unroll
  for (int r = 0; r < 8; r++) {
    bs0t0[r] = bih0[r + go] + bhh0[r + go];
    bs0t1[r] = bih0[16 + r + go] + bhh0[16 + r + go];
    bs1t0[r] = bih1[r + go] + bhh1[r + go];
    bs1t1[r] = bih1[16 + r + go] + bhh1[16 + r + go];
    bs2t0[r] = bih2[r + go] + bhh2[r + go];
    bs2t1[r] = bih2[16 + r + go] + bhh2[16 + r + go];
  }
  // layer-0 input weights (W_ih0 is [32,1])
  float wx0[8], wx1[8];
#pragma unroll
  for (int r = 0; r < 8; r++) {
    wx0[r] = Wih0[r + go];
    wx1[r] = Wih0[16 + r + go];
  }

  float B0[4] = {0.f, 0.f, 0.f, 0.f};
  float B1[4] = {0.f, 0.f, 0.f, 0.f};
  float B2[4] = {0.f, 0.f, 0.f, 0.f};
  float C0[8] = {0.f}, C1[8] = {0.f}, C2[8] = {0.f};
  float Hcur[8];

  const float* xrow = x + (size_t)(b0 + bl) * Tt;
  float xv = xrow[0];  // software-pipelined input load

  for (int t = 0; t < Tt; t++) {
    // Unconditional prefetch with clamped (in-bounds) index: keeps the CFG
    // straight-line so the waitcnt pass can use distance-1 waits, and the
    // load for step t+1 stays in flight across this whole step's work.
    int tn = t + 1 < Tt ? t + 1 : Tt - 1;
    float xv_next = xrow[tn];

    v8f d0, d1;
    // layer 0: xg = x*W_ih0 + bias, then + Whh0 @ h0^T
#pragma unroll
    for (int r = 0; r < 8; r++) {
      d0[r] = xv * wx0[r] + bs0t0[r];
      d1[r] = xv * wx1[r] + bs0t1[r];
    }
    d0 = mm8(aH0 + 0, B0, d0);
    d1 = mm8(aH0 + 4, B0, d1);
    cell(d0, d1, lo, C0, B0, Hcur);   // B0 := new h0 (input to layer 1)

    // layer 1
#pragma unroll
    for (int r = 0; r < 8; r++) { d0[r] = bs1t0[r]; d1[r] = bs1t1[r]; }
    d0 = mm8(aI1 + 0, B0, d0);
    d0 = mm8(aH1 + 0, B1, d0);
    d1 = mm8(aI1 + 4, B0, d1);
    d1 = mm8(aH1 + 4, B1, d1);
    cell(d0, d1, lo, C1, B1, Hcur);

    // layer 2
#pragma unroll
    for (int r = 0; r < 8; r++) { d0[r] = bs2t0[r]; d1[r] = bs2t1[r]; }
    d0 = mm8(aI2 + 0, B1, d0);
    d0 = mm8(aH2 + 0, B2, d0);
    d1 = mm8(aI2 + 4, B1, d1);
    d1 = mm8(aH2 + 4, B2, d1);
    cell(d0, d1, lo, C2, B2, Hcur);

    if (lo) {
      float4 v0 = {Hcur[0], Hcur[1], Hcur[2], Hcur[3]};
      float4 v1 = {Hcur[4], Hcur[5], Hcur[6], Hcur[7]};
      float* p = out + (size_t)(b0 + bl) * (Tt * Hh) + t * Hh;
      *(float4*)(p) = v0;
      *(float4*)(p + 4) = v1;
    }
    xv = xv_next;
  }
}

// Attention + fc + final FC. One wave per batch element.
// S = out @ out^T via WMMA; softmax over s; fcv[t] = softmax . v + fc_b;
// y[b,o] = fcv . ffc_w[o,:] + ffc_b[o].
__global__ __launch_bounds__(256) void attn_kernel(
    const float* __restrict__ out,   // [B,90,8]
    const float* __restrict__ fcw,   // [8]
    const float* __restrict__ fcb,   // [1]
    const float* __restrict__ ffcw,  // [30,90]
    const float* __restrict__ ffcb,  // [30]
    float* __restrict__ y) {         // [B,30]
  __shared__ float s_ffcw[OUTL * Tt];
  __shared__ float s_fcv[8][96];

  for (int i = threadIdx.x; i < OUTL * Tt; i += 256) s_ffcw[i] = ffcw[i];
  __syncthreads();

  const int lane = threadIdx.x & 31;
  const int wave = threadIdx.x >> 5;
  const int b = blockIdx.x * 8 + wave;
  const bool lo = lane < 16;
  const int rl = lane & 15;
  const int kof = lo ? 0 : 2;
  const float* ob = out + (size_t)b * (Tt * Hh);

  // 6 row-chunks of out; identical layout serves as both A (M=row,K=h) and B (K=h,N=row)
  float ch[6][4];
#pragma unroll
  for (int j = 0; j < 6; j++) {
    int row = j * 16 + rl;
    bool valid = row < Tt;
#pragma unroll
    for (int c = 0; c < 2; c++)
#pragma unroll
      for (int u = 0; u < 2; u++)
        ch[j][c * 2 + u] = valid ? ob[row * Hh + (c * 4 + u + kof)] : 0.0f;
  }

  // v[s] = out[s,:] . fc_w ; each lane holds v for its row via cross-half combine
  float fw[4];
#pragma unroll
  for (int c = 0; c < 2; c++)
#pragma unroll
    for (int u = 0; u < 2; u++) fw[c * 2 + u] = fcw[c * 4 + u + kof];
  float vv[6];
#pragma unroll
  for (int j = 0; j < 6; j++) {
    float p = ch[j][0] * fw[0] + ch[j][1] * fw[1] + ch[j][2] * fw[2] + ch[j][3] * fw[3];
    vv[j] = p + __shfl_xor(p, 16, 32);
  }

  const float fcbias = fcb[0];
  float fcv[6][8];
#pragma unroll
  for (int jt = 0; jt < 6; jt++) {
    v8f S[6];
#pragma unroll
    for (int js = 0; js < 6; js++) {
      v8f c = {0.f, 0.f, 0.f, 0.f, 0.f, 0.f, 0.f, 0.f};
      c = wmma4(ch[jt][2], ch[jt][3], ch[js][2], ch[js][3], c);
      c = wmma4(ch[jt][0], ch[jt][1], ch[js][0], ch[js][1], c);
      S[js] = c;
    }
    // mask padded keys s>=90 (s-tile 5, lanes rl>=10)
    if (rl >= 10) {
#pragma unroll
      for (int r = 0; r < 8; r++) S[5][r] = -3.0e38f;
    }
#pragma unroll
    for (int r = 0; r < 8; r++) {
      float m = S[0][r];
#pragma unroll
      for (int js = 1; js < 6; js++) m = fmaxf(m, S[js][r]);
#pragma unroll
      for (int d = 1; d < 16; d <<= 1) m = fmaxf(m, __shfl_xor(m, d, 32));
      float se = 0.f, sv = 0.f;
#pragma unroll
      for (int js = 0; js < 6; js++) {
        float e = __expf(S[js][r] - m);
        se += e;
        sv += e * vv[js];
      }
#pragma unroll
      for (int d = 1; d < 16; d <<= 1) {
        se += __shfl_xor(se, d, 32);
        sv += __shfl_xor(sv, d, 32);
      }
      fcv[jt][r] = sv * __builtin_amdgcn_rcpf(se) + fcbias;
    }
  }

  // stash fcv[t] (t = 16*jt + (lo?0:8) + r) in LDS, then final 90->30 FC
  if (lane == 0 || lane == 16) {
    int off = lo ? 0 : 8;
#pragma unroll
    for (int jt = 0; jt < 6; jt++)
#pragma unroll
      for (int r = 0; r < 8; r++) s_fcv[wave][16 * jt + off + r] = fcv[jt][r];
  }
  __syncthreads();

  if (lane < OUTL) {
    float acc = ffcb[lane];
    for (int t = 0; t < Tt; t++) acc += s_fcv[wave][t] * s_ffcw[lane * Tt + t];
    y[(size_t)b * OUTL + lane] = acc;
  }
}

extern "C" void kernel_launch(void* const* d_in, const int* in_sizes, int n_in,
                              void* d_out, int out_size, void* d_ws, size_t ws_size,
                              hipStream_t stream) {
  (void)in_sizes; (void)n_in; (void)out_size; (void)ws_size;
  const float* x    = (const float*)d_in[0];
  const float* Wih0 = (const float*)d_in[1];
  const float* Whh0 = (const float*)d_in[2];
  const float* bih0 = (const float*)d_in[3];
  const float* bhh0 = (const float*)d_in[4];
  const float* Wih1 = (const float*)d_in[5];
  const float* Whh1 = (const float*)d_in[6];
  const float* bih1 = (const float*)d_in[7];
  const float* bhh1 = (const float*)d_in[8];
  const float* Wih2 = (const float*)d_in[9];
  const float* Whh2 = (const float*)d_in[10];
  const float* bih2 = (const float*)d_in[11];
  const float* bhh2 = (const float*)d_in[12];
  const float* fcw  = (const float*)d_in[13];
  const float* fcb  = (const float*)d_in[14];
  const float* ffcw = (const float*)d_in[15];
  const float* ffcb = (const float*)d_in[16];

  float* outbuf = (float*)d_ws;  // [B,90,8] f32 = 47.2 MB scratch

  lstm3_kernel<<<dim3(Bb / 16 / 8), dim3(256), 0, stream>>>(
      x, Wih0, Whh0, bih0, bhh0, Wih1, Whh1, bih1, bhh1, Wih2, Whh2, bih2, bhh2,
      outbuf);
  attn_kernel<<<dim3(Bb / 8), dim3(256), 0, stream>>>(
      outbuf, fcw, fcb, ffcw, ffcb, (float*)d_out);
}